// Encoder_17282948399460
// MI455X (gfx1250) — compile-verified
//
#include <hip/hip_runtime.h>

// ---------------------------------------------------------------------------
// Density-based sampler, fused for MI455X (gfx1250, wave32, WMMA).
//   features: (8, 4096, 60) f32, pos: (8, 4096, 3) f32, cam_ids: (8, 4096) i32
//   out = [sampled_features (8,819,60) | sampled_pos (8,819,3) | sampled_cam (8,819)]
//
// Roofline: the naive path writes/reads a 512 MB (B,N,N) distance tensor
// (~1 GB HBM traffic); here it never leaves the WGP. Each wave owns a 16-row
// tile, streams all column tiles through V_WMMA_F32_16X16X4_F32 (full f32,
// matching the reference Gram-trick numerics), and keeps the per-row
// 8-smallest *squared* distances in registers — sqrt is applied only to the
// 8 survivors per row. Features (8 MB) live in the 192 MB L2 across all
// 256 row-tile waves per batch.
// ---------------------------------------------------------------------------

typedef float v2f __attribute__((ext_vector_type(2)));
typedef float v8f __attribute__((ext_vector_type(8)));

#define B_   8
#define N_   4096
#define C_   60
#define KNN  8
#define M_   819          // N_ / 5
#define FBIG 3.402823466e38f

// --------------------------- kernel 0: |x|^2 per point ----------------------
__global__ __launch_bounds__(256)
void sq_kernel(const float* __restrict__ feat, float* __restrict__ sq) {
    int gid = blockIdx.x * blockDim.x + threadIdx.x;        // [0, B*N)
    if (gid >= B_ * N_) return;
    const float* row = feat + (size_t)gid * C_;
    float s = 0.f;
#pragma unroll
    for (int c = 0; c < C_; ++c) s = fmaf(row[c], row[c], s);
    sq[gid] = s;
}

// branchless sorted insert of c into ascending 8-list:
//   t[0]=min(s0,c); t[i]=min(s[i], max(c, s[i-1]))
__device__ __forceinline__ void knn_insert(float (&knn)[KNN], float c) {
#pragma unroll
    for (int q = 0; q < KNN; ++q) {
        float sv = knn[q];
        knn[q] = fminf(sv, c);
        c      = fmaxf(c, sv);
    }
}

// ------------------ kernel 1: fused WMMA gram -> knn density ----------------
// one wave (32 threads) per (batch, 16-row tile); grid = B * 256
__global__ __launch_bounds__(32)
void density_kernel(const float* __restrict__ feat,
                    const float* __restrict__ sq,
                    float* __restrict__ density) {
    // ping-pong 16x32 d^2 staging tiles + merge scratch
    __shared__ float ldsTile[2][16 * 32];
    __shared__ float ldsMerge[16 * KNN];

    const int wid  = blockIdx.x;         // 0 .. B*256-1
    const int b    = wid >> 8;
    const int i0   = (wid & 255) << 4;   // row-tile base
    const int lane = threadIdx.x;        // 0..31
    const int half = lane >> 4;          // 0 | 1
    const int l15  = lane & 15;

    const float* fb  = feat + (size_t)b * N_ * C_;
    const float* sqb = sq + b * N_;

    // A tile (rows i0..i0+15), resident: 15 k-blocks of float2.
    // f32 16x16x4 A layout: lane l -> row (l&15); lanes 0-15 hold K={0,1},
    // lanes 16-31 hold K={2,3} of each 4-wide k-block.
    v2f a[15];
#pragma unroll
    for (int kb = 0; kb < 15; ++kb) {
        const float* p = fb + (size_t)(i0 + l15) * C_ + 4 * kb + 2 * half;
        a[kb].x = p[0];
        a[kb].y = p[1];
    }

    // |x|^2 for the 8 rows this lane-half sees in the C tile (rows v + 8*half)
    float sA[8];
#pragma unroll
    for (int v = 0; v < 8; ++v) sA[v] = sqb[i0 + v + 8 * half];

    // running 8-smallest squared distances (ascending), per (row=l15, half)
    float knn[KNN];
#pragma unroll
    for (int q = 0; q < KNN; ++q) knn[q] = FBIG;

    int buf = 0;
    for (int j0 = 0; j0 < N_; j0 += 32, buf ^= 1) {
        // two B tiles (cols j0..j0+15 and j0+16..j0+31); B[k][n] = F[col, k]
        v2f bt0[15], bt1[15];
#pragma unroll
        for (int kb = 0; kb < 15; ++kb) {
            const float* p0 = fb + (size_t)(j0 + l15) * C_ + 4 * kb + 2 * half;
            const float* p1 = p0 + 16 * C_;
            bt0[kb].x = p0[0]; bt0[kb].y = p0[1];
            bt1[kb].x = p1[0]; bt1[kb].y = p1[1];
        }

        v8f acc0 = {};
        v8f acc1 = {};
#pragma unroll
        for (int kb = 0; kb < 15; ++kb) {
            acc0 = __builtin_amdgcn_wmma_f32_16x16x4_f32(
                false, a[kb], false, bt0[kb], (short)0, acc0, false, false);
            acc1 = __builtin_amdgcn_wmma_f32_16x16x4_f32(
                false, a[kb], false, bt1[kb], (short)0, acc1, false, false);
        }

        const float sqB0 = sqb[j0 + l15];        // col n = l15      (tile 0)
        const float sqB1 = sqb[j0 + 16 + l15];   // col n = l15 + 16 (tile 1)

        // d^2 = max(|xi|^2 + |xj|^2 - 2*gram, 0); stage both tiles into LDS
        float* tile = ldsTile[buf];
#pragma unroll
        for (int v = 0; v < 8; ++v) {
            float d0 = fmaxf(fmaf(-2.f, acc0[v], sA[v] + sqB0), 0.f);
            float d1 = fmaxf(fmaf(-2.f, acc1[v], sA[v] + sqB1), 0.f);
            tile[(v + 8 * half) * 32 + l15]      = d0;
            tile[(v + 8 * half) * 32 + 16 + l15] = d1;
        }
        __syncthreads();   // single-wave WG: cheap; orders write -> read

        // lane owns row r=l15, columns [16*half, 16*half+16): 4 x float4
        const float4* rowp =
            (const float4*)&tile[l15 * 32 + 16 * half];
#pragma unroll
        for (int t = 0; t < 4; ++t) {
            float4 c4 = rowp[t];
            knn_insert(knn, c4.x);
            knn_insert(knn, c4.y);
            knn_insert(knn, c4.z);
            knn_insert(knn, c4.w);
        }
        // ping-pong buffer: next iteration writes the other tile, and the
        // following __syncthreads fences these reads before buffer reuse.
    }

    // merge the two half-row lists -> mean of sqrt of the 8 smallest d^2
    if (half == 1) {
#pragma unroll
        for (int t = 0; t < KNN; ++t) ldsMerge[l15 * KNN + t] = knn[t];
    }
    __syncthreads();
    if (half == 0) {
#pragma unroll
        for (int t = 0; t < KNN; ++t) knn_insert(knn, ldsMerge[l15 * KNN + t]);
        float sum = 0.f;
#pragma unroll
        for (int q = 0; q < KNN; ++q) sum += sqrtf(knn[q]);
        density[b * N_ + i0 + l15] = sum * (1.0f / KNN);
    }
}

// --------- kernel 2: per-batch bitonic sort (value desc, idx asc) -----------
__global__ __launch_bounds__(1024)
void topm_kernel(const float* __restrict__ density, int* __restrict__ sampled) {
    __shared__ float val[N_];
    __shared__ int   idx[N_];
    const int b   = blockIdx.x;
    const int tid = threadIdx.x;

    for (int i = tid; i < N_; i += 1024) { val[i] = density[b * N_ + i]; idx[i] = i; }
    __syncthreads();

    for (int k = 2; k <= N_; k <<= 1) {
        for (int j = k >> 1; j > 0; j >>= 1) {
            for (int i = tid; i < N_; i += 1024) {
                int ixj = i ^ j;
                if (ixj > i) {
                    float v1 = val[i], v2 = val[ixj];
                    int   x1 = idx[i], x2 = idx[ixj];
                    // before(a,b): a.v > b.v  ||  (a.v == b.v && a.i < b.i)
                    bool b21 = (v2 > v1) || (v2 == v1 && x2 < x1);
                    bool b12 = (v1 > v2) || (v1 == v2 && x1 < x2);
                    bool up  = ((i & k) == 0);        // descending block
                    bool sw  = up ? b21 : b12;
                    if (sw) {
                        val[i] = v2; val[ixj] = v1;
                        idx[i] = x2; idx[ixj] = x1;
                    }
                }
            }
            __syncthreads();
        }
    }
    for (int m = tid; m < M_; m += 1024) sampled[b * M_ + m] = idx[m];
}

// ----------------------- kernel 3: gather the outputs -----------------------
// one 64-thread block per (b, m): t<60 features, t in [60,63) pos, t==63 cam
__global__ __launch_bounds__(64)
void gather_kernel(const float* __restrict__ feat, const float* __restrict__ pos,
                   const int* __restrict__ cam, const int* __restrict__ sampled,
                   float* __restrict__ out) {
    const int bm  = blockIdx.x;          // 0 .. B*M-1  (bm = b*M + m)
    const int b   = bm / M_;
    const int t   = threadIdx.x;
    const int src = sampled[bm];

    const size_t posBase = (size_t)B_ * M_ * C_;
    const size_t camBase = posBase + (size_t)B_ * M_ * 3;

    if (t < C_) {
        out[(size_t)bm * C_ + t] = feat[((size_t)b * N_ + src) * C_ + t];
    } else if (t < 63) {
        out[posBase + (size_t)bm * 3 + (t - 60)] = pos[((size_t)b * N_ + src) * 3 + (t - 60)];
    } else {
        out[camBase + bm] = (float)cam[b * N_ + src];
    }
}

// ---------------------------------------------------------------------------
extern "C" void kernel_launch(void* const* d_in, const int* in_sizes, int n_in,
                              void* d_out, int out_size, void* d_ws, size_t ws_size,
                              hipStream_t stream) {
    const float* feat = (const float*)d_in[0];
    const float* pos  = (const float*)d_in[1];
    const int*   cam  = (const int*)d_in[2];
    float* out = (float*)d_out;

    // workspace layout: sq (B*N f32) | density (B*N f32) | sampled (B*M i32)
    float* sq      = (float*)d_ws;
    float* density = sq + (size_t)B_ * N_;
    int*   sampled = (int*)(density + (size_t)B_ * N_);

    sq_kernel<<<(B_ * N_ + 255) / 256, 256, 0, stream>>>(feat, sq);
    density_kernel<<<B_ * (N_ / 16), 32, 0, stream>>>(feat, sq, density);
    topm_kernel<<<B_, 1024, 0, stream>>>(density, sampled);
    gather_kernel<<<B_ * M_, 64, 0, stream>>>(feat, pos, cam, sampled, out);
}